// SinkhornMatchingLoss_68934225101148
// MI455X (gfx1250) — compile-verified
//
#include <hip/hip_runtime.h>

typedef __attribute__((ext_vector_type(2))) float v2f;
typedef __attribute__((ext_vector_type(8))) float v8f;

#define SK_TAU   0.05f
#define SK_EPS   1e-8f
#define SK_ITER  5
#define SK_N     128
#define SK_D     64

__global__ void __launch_bounds__(256, 1)
sinkhorn_zero_kernel(float* out) {
    if (threadIdx.x == 0 && blockIdx.x == 0) out[0] = 0.0f;
}

__global__ void __launch_bounds__(256, 1)
sinkhorn_loss_kernel(const float* __restrict__ src,
                     const float* __restrict__ tgt,
                     float* __restrict__ out,
                     int num_frames)
{
    __shared__ float lsrc[SK_N * SK_D];   // 32 KB
    __shared__ float ltgt[SK_N * SK_D];   // 32 KB
    __shared__ float Cm[SK_N * SK_N];     // 64 KB
    __shared__ float Km[SK_N * SK_N];     // 64 KB
    __shared__ float x2[SK_N];
    __shared__ float y2[SK_N];
    __shared__ float uu[SK_N];
    __shared__ float vv[SK_N];
    __shared__ float red[256];

    const int tid = threadIdx.x;
    const int bt  = blockIdx.x;

    const float* gs = src + (size_t)bt * SK_N * SK_D;
    const float* gt = tgt + (size_t)bt * SK_N * SK_D;

    // ---- Stage frame to LDS with async-to-LDS b128 (ASYNCcnt path) ----
    // Each lane moves 16B/op directly global->LDS, no VGPR round trip.
    {
        const unsigned int lsrc_off = (unsigned int)(uintptr_t)lsrc;
        const unsigned int ltgt_off = (unsigned int)(uintptr_t)ltgt;
        #pragma unroll
        for (int i = 0; i < SK_N * SK_D / 4 / 256; ++i) {   // 8 iters
            const int e = (i * 256 + tid) * 16;             // byte offset
            unsigned int      ls = lsrc_off + (unsigned int)e;
            unsigned int      lt = ltgt_off + (unsigned int)e;
            unsigned long long as = (unsigned long long)(uintptr_t)((const char*)gs + e);
            unsigned long long at = (unsigned long long)(uintptr_t)((const char*)gt + e);
            asm volatile("global_load_async_to_lds_b128 %0, %1, off"
                         :: "v"(ls), "v"(as) : "memory");
            asm volatile("global_load_async_to_lds_b128 %0, %1, off"
                         :: "v"(lt), "v"(at) : "memory");
        }
        asm volatile("s_wait_asynccnt 0" ::: "memory");
    }
    __syncthreads();

    // ---- Row squared norms ----
    if (tid < 2 * SK_N) {
        const float* base = (tid < SK_N) ? (lsrc + tid * SK_D)
                                         : (ltgt + (tid - SK_N) * SK_D);
        float s = 0.0f;
        for (int k = 0; k < SK_D; ++k) { float a = base[k]; s += a * a; }
        if (tid < SK_N) x2[tid] = s; else y2[tid - SK_N] = s;
    }
    if (tid < SK_N) vv[tid] = 1.0f;
    __syncthreads();

    // ---- Cost matrix via fp32 WMMA: xy = src @ tgt^T, tile 16x16, K=64 ----
    {
        const int wave  = tid >> 5;
        const int lane  = tid & 31;
        const int mn    = lane & 15;            // A row / B col within tile
        const int khalf = (lane >> 4) << 1;     // 0 or 2: K-pair selector
        const int rowHi = (lane >> 4) << 3;     // 0 or 8: C/D row half

        for (int t = 0; t < 8; ++t) {
            const int tile = wave * 8 + t;      // 64 tiles, 8 waves x 8
            const int ti = tile >> 3;
            const int tj = tile & 7;
            const float* arow = lsrc + (ti * 16 + mn) * SK_D;
            const float* brow = ltgt + (tj * 16 + mn) * SK_D;

            v8f c = {};
            #pragma unroll
            for (int ks = 0; ks < 16; ++ks) {
                const int k0 = ks * 4 + khalf;
                v2f a = *(const v2f*)(arow + k0);
                v2f b = *(const v2f*)(brow + k0);
                c = __builtin_amdgcn_wmma_f32_16x16x4_f32(
                        /*neg_a=*/false, a, /*neg_b=*/false, b,
                        /*c_mod=*/(short)0, c,
                        /*reuse_a=*/false, /*reuse_b=*/false);
            }

            const int col = tj * 16 + mn;
            #pragma unroll
            for (int r = 0; r < 8; ++r) {
                const int row = ti * 16 + rowHi + r;
                float xy = c[r];
                float d2 = x2[row] + y2[col] - 2.0f * xy;
                float dd = __builtin_amdgcn_sqrtf(fmaxf(d2, 0.0f)); // raw v_sqrt_f32
                Cm[row * SK_N + col] = dd;
                Km[row * SK_N + col] = __expf(dd * (-1.0f / SK_TAU));
            }
        }
    }
    __syncthreads();

    // ---- Sinkhorn iterations: each row/col split across 2 threads ----
    const int rc   = tid & (SK_N - 1);
    const int half = tid >> 7;               // 0 or 1
    for (int it = 0; it < SK_ITER; ++it) {
        // u = mu / (K v + eps)
        {
            const float* kr = Km + rc * SK_N + half * 64;
            const float* vh = vv + half * 64;
            float s = 0.0f;
            #pragma unroll 8
            for (int m = 0; m < 64; ++m) s += kr[m] * vh[m];
            red[tid] = s;
        }
        __syncthreads();
        if (tid < SK_N) uu[tid] = (1.0f / SK_N) / (red[tid] + red[tid + 128] + SK_EPS);
        __syncthreads();
        // v = nu / (K^T u + eps)
        {
            float s = 0.0f;
            #pragma unroll 8
            for (int n = 0; n < 64; ++n)
                s += Km[(half * 64 + n) * SK_N + rc] * uu[half * 64 + n];
            red[tid] = s;
        }
        __syncthreads();
        if (tid < SK_N) vv[tid] = (1.0f / SK_N) / (red[tid] + red[tid + 128] + SK_EPS);
        __syncthreads();
    }

    // ---- frame_loss = sum(u_n * K_nm * v_m * C_nm) ----
    float acc = 0.0f;
    int e = tid * 64;
    #pragma unroll 4
    for (int i = 0; i < 64; ++i, ++e) {
        const int row = e >> 7;
        const int col = e & (SK_N - 1);
        acc += uu[row] * Km[e] * vv[col] * Cm[e];
    }
    red[tid] = acc;
    __syncthreads();
    for (int s = 128; s > 0; s >>= 1) {
        if (tid < s) red[tid] += red[tid + s];
        __syncthreads();
    }
    if (tid == 0) {
        atomicAdd(out, red[0] * (1.0f / (float)num_frames));
    }
}

extern "C" void kernel_launch(void* const* d_in, const int* in_sizes, int n_in,
                              void* d_out, int out_size, void* d_ws, size_t ws_size,
                              hipStream_t stream) {
    (void)n_in; (void)out_size; (void)d_ws; (void)ws_size;
    const float* src = (const float*)d_in[0];
    const float* tgt = (const float*)d_in[1];
    float* out = (float*)d_out;

    const int num_frames = in_sizes[0] / (SK_N * SK_D);   // B*T = 2048

    sinkhorn_zero_kernel<<<1, 256, 0, stream>>>(out);
    sinkhorn_loss_kernel<<<num_frames, 256, 0, stream>>>(src, tgt, out, num_frames);
}